// NodeVarGraphConvolutionLayer_20109036880595
// MI455X (gfx1250) — compile-verified
//
#include <hip/hip_runtime.h>
#include <hip/hip_bf16.h>
#include <math.h>

// Problem constants (from reference setup_inputs)
#define B_   16
#define N_   1024
#define CIN_ 64
#define C_   64
#define KT_  3      // number of polynomial terms beyond P0

typedef __attribute__((ext_vector_type(2))) float v2f;
typedef __attribute__((ext_vector_type(8))) float v8f;

// ---------------------------------------------------------------------------
// Kernel 1: P0[b,n] = sum_ch X[b,n,ch]   (X row is 64 contiguous floats)
// ---------------------------------------------------------------------------
__global__ void xsum_kernel(const float* __restrict__ X, float* __restrict__ P0) {
  const int idx = blockIdx.x * blockDim.x + threadIdx.x;   // 0 .. B*N-1
  const float4* row = (const float4*)(X + (size_t)idx * CIN_);
  float s = 0.f;
#pragma unroll
  for (int i = 0; i < CIN_ / 4; ++i) {
    float4 v = row[i];
    s += v.x + v.y + v.z + v.w;
  }
  P0[idx] = s;
}

// ---------------------------------------------------------------------------
// Kernel 2: batched mat-vec  pout[b,n] = sum_m A[b,n,m] * pin[b,m]
// One block = one 16-row slab of one batch. 8 waves split the K=1024
// contraction into 128-wide segments; each wave runs 32 x
// V_WMMA_F32_16X16X4_F32 with the vector chunk replicated across the 16
// B-columns, partial sums are reduced through LDS.
// ---------------------------------------------------------------------------
__global__ void __launch_bounds__(256)
matvec_wmma_kernel(const float* __restrict__ A,
                   const float* __restrict__ pin,
                   float* __restrict__ pout) {
  __shared__ float red[8][16];

  const int lane = threadIdx.x & 31;
  const int wave = threadIdx.x >> 5;        // 0..7 -> K segment
  const int slab = blockIdx.x;              // 0 .. B*(N/16)-1
  const int b    = slab >> 6;               // N/16 = 64 slabs per batch
  const int n0   = (slab & 63) << 4;        // first output row of slab
  const int half = lane >> 4;               // 0 | 1 (K-pair select)
  const int row  = lane & 15;               // M within tile

  const int kseg = wave * (N_ / 8);         // 128-wide K segment
  const float* Ap = A + (size_t)b * N_ * N_ + (size_t)(n0 + row) * N_
                      + kseg + 2 * half;
  const float* pp = pin + b * N_ + kseg + 2 * half;

  v8f acc = {};
#pragma unroll 4
  for (int k = 0; k < N_ / 8; k += 4) {
    // A-tile 16x4 f32: VGPR0 = K=2h, VGPR1 = K=2h+1 per lane (contiguous pair)
    v2f a = *(const v2f*)(Ap + k);
    // B-tile 4x16 f32 with all 16 columns equal to the vector chunk
    v2f p = *(const v2f*)(pp + k);
    acc = __builtin_amdgcn_wmma_f32_16x16x4_f32(
        /*neg_a=*/false, a, /*neg_b=*/false, p,
        /*c_mod=*/(short)0, acc, /*reuse_a=*/false, /*reuse_b=*/false);
  }

  // D layout: VGPR v -> row v (lanes 0-15) / row v+8 (lanes 16-31); all
  // columns identical, so lane 0 and lane 16 carry the 16 row results.
  if (row == 0) {
    const int base = half * 8;
#pragma unroll
    for (int r = 0; r < 8; ++r) red[wave][base + r] = acc[r];
  }
  __syncthreads();

  if (threadIdx.x < 16) {
    float s = 0.f;
#pragma unroll
    for (int w = 0; w < 8; ++w) s += red[w][threadIdx.x];
    pout[b * N_ + n0 + threadIdx.x] = s;   // coalesced 64B store
  }
}

// ---------------------------------------------------------------------------
// Kernel 3: transpose taps  ht[i,n,c] = h[i,c,n]  (1 MB, stays in L2)
// ---------------------------------------------------------------------------
__global__ void htrans_kernel(const float* __restrict__ h, float* __restrict__ ht) {
  const int idx  = blockIdx.x * blockDim.x + threadIdx.x;  // (i*N + n)*C + c
  const int c    = idx & (C_ - 1);
  const int rest = idx >> 6;
  const int n    = rest & (N_ - 1);
  const int i    = rest >> 10;
  ht[idx] = h[((size_t)i * C_ + c) * N_ + n];
}

// ---------------------------------------------------------------------------
// Kernel 4: Y[b,n,c] = sum_i ht[i,n,c] * P[i,b,n]; LayerNorm over C; tanh.
// One wave32 per (b,n); 2 channels per lane; shfl_xor wave reductions.
// ---------------------------------------------------------------------------
__global__ void __launch_bounds__(256)
final_kernel(const float* __restrict__ P,     // [K+1][B*N]
             const float* __restrict__ ht,    // [K+1][N][C]
             const float* __restrict__ gamma, // [C]
             const float* __restrict__ beta,  // [C]
             float* __restrict__ out) {       // [B,N,C]
  const int lane  = threadIdx.x & 31;
  const int gwave = (blockIdx.x * blockDim.x + threadIdx.x) >> 5;  // 0..B*N-1
  const int b = gwave >> 10;
  const int n = gwave & (N_ - 1);

  float y0 = 0.f, y1 = 0.f;
#pragma unroll
  for (int i = 0; i <= KT_; ++i) {
    float p = P[i * (B_ * N_) + b * N_ + n];             // wave-uniform
    const float* hrow = ht + ((size_t)i * N_ + n) * C_;  // contiguous 256B
    y0 = fmaf(p, hrow[lane],      y0);
    y1 = fmaf(p, hrow[lane + 32], y1);
  }

  float s1 = y0 + y1;
  float s2 = fmaf(y0, y0, y1 * y1);
#pragma unroll
  for (int m = 16; m >= 1; m >>= 1) {
    s1 += __shfl_xor(s1, m, 32);
    s2 += __shfl_xor(s2, m, 32);
  }
  const float mu   = s1 * (1.f / C_);
  const float var  = s2 * (1.f / C_) - mu * mu;   // biased var, matches jnp.var
  const float rstd = rsqrtf(var + 1e-5f);

  float* dst = out + (size_t)gwave * C_;
  dst[lane]      = tanhf(fmaf((y0 - mu) * rstd, gamma[lane],      beta[lane]));
  dst[lane + 32] = tanhf(fmaf((y1 - mu) * rstd, gamma[lane + 32], beta[lane + 32]));
}

// ---------------------------------------------------------------------------
extern "C" void kernel_launch(void* const* d_in, const int* in_sizes, int n_in,
                              void* d_out, int out_size, void* d_ws, size_t ws_size,
                              hipStream_t stream) {
  const float* A     = (const float*)d_in[0];   // [B,N,N]
  const float* X     = (const float*)d_in[1];   // [B,N,Cin]
  const float* h     = (const float*)d_in[2];   // [K+1,C,N]
  const float* gamma = (const float*)d_in[3];   // [C]
  const float* beta  = (const float*)d_in[4];   // [C]
  float* out = (float*)d_out;                   // [B,N,C]

  // Workspace: P[(K+1)][B*N] then ht[(K+1)][N][C]  (~1.25 MB total)
  float* P  = (float*)d_ws;
  float* ht = P + (size_t)(KT_ + 1) * B_ * N_;

  // P0 = X.sum(-1)
  hipLaunchKernelGGL(xsum_kernel, dim3(B_ * N_ / 256), dim3(256), 0, stream, X, P);
  // tap transpose (independent of mat-vec chain)
  hipLaunchKernelGGL(htrans_kernel, dim3((KT_ + 1) * N_ * C_ / 256), dim3(256), 0,
                     stream, h, ht);
  // sequential power chain: P[i+1] = A @ P[i]
  for (int i = 0; i < KT_; ++i) {
    hipLaunchKernelGGL(matvec_wmma_kernel, dim3(B_ * (N_ / 16)), dim3(256), 0,
                       stream, A, P + i * (B_ * N_), P + (i + 1) * (B_ * N_));
  }
  // combine + LayerNorm + tanh: one wave per (b,n)
  hipLaunchKernelGGL(final_kernel, dim3((B_ * N_ * 32) / 256), dim3(256), 0,
                     stream, P, ht, gamma, beta, out);
}